// SRN_criterion_2104533975230
// MI455X (gfx1250) — compile-verified
//
#include <hip/hip_runtime.h>
#include <cstdint>

// Problem constants (match reference: B=16, P=65536, T=64)
#define BN   16
#define PN   65536
#define TN   64
#define BLK  256
#define NBLK (BN * (PN / BLK))   // 4096 blocks in the loss pass

typedef __attribute__((ext_vector_type(2))) float v2f;
typedef __attribute__((ext_vector_type(8))) float v8f;
typedef __attribute__((ext_vector_type(4))) int   v4i;

// ---------------------------------------------------------------------------
// Wave32 sum reduction via V_WMMA_F32_16X16X4_F32.
// A (16x4 f32): lane L<16 holds A[M=L][K=0..1] in 2 VGPRs, lane L>=16 holds
// K=2..3.  Put x in K-slot 0, zero in the other -> rowsum(M) = x[M] + x[M+16].
// B (4x16) = all ones -> D[M][N] = rowsum(M) broadcast over N.
// C/D layout: lanes 0..15 VGPR r = rowsum(r); lanes 16..31 VGPR r = rowsum(r+8).
// Per-lane sum of the 8 D VGPRs at lane 0 covers rows 0..7, lane 16 rows 8..15.
// total = readlane(s,0) + readlane(s,16)  (v_readlane, no LDS traffic).
// ---------------------------------------------------------------------------
__device__ __forceinline__ float wave_sum32(float x) {
  v2f a; a.x = x;    a.y = 0.0f;
  v2f b; b.x = 1.0f; b.y = 1.0f;
  v8f c = {};
  c = __builtin_amdgcn_wmma_f32_16x16x4_f32(
      /*neg_a=*/false, a, /*neg_b=*/false, b,
      /*c_mod=*/(short)0, c, /*reuse_a=*/false, /*reuse_b=*/false);
  float s = ((c[0] + c[1]) + (c[2] + c[3])) + ((c[4] + c[5]) + (c[6] + c[7]));
  float s0  = __int_as_float(__builtin_amdgcn_readlane(__float_as_int(s), 0));
  float s16 = __int_as_float(__builtin_amdgcn_readlane(__float_as_int(s), 16));
  return s0 + s16;
}

// ---------------------------------------------------------------------------
// Stage the 64 truth boxes (float4) into LDS via the gfx1250 async-to-LDS
// path (ASYNCcnt-tracked) when the toolchain exposes the builtins.
// Builtin signature (verified by hipcc diagnostic):
//   void __builtin_amdgcn_global_load_async_to_lds_b128(
//       v4i addrspace(1)* gsrc, v4i addrspace(3)* ldst, imm int off, imm int cpol)
// uintptr_t round-trip: AS1 ptr == low 64 bits of generic global ptr; AS3 ptr
// == low 32 bits of generic LDS ptr on amdgcn.
// ---------------------------------------------------------------------------
#if __has_builtin(__builtin_amdgcn_global_load_async_to_lds_b128) && \
    __has_builtin(__builtin_amdgcn_s_wait_asynccnt)
#define HAVE_ASYNC_LDS 1
#else
#define HAVE_ASYNC_LDS 0
#endif

__device__ __forceinline__ void stage_truths(const float4* __restrict__ g,
                                             float4* s, int tid) {
#if HAVE_ASYNC_LDS
  if (tid < TN) {
    __attribute__((address_space(1))) v4i* gp =
        (__attribute__((address_space(1))) v4i*)(uintptr_t)(g + tid);
    __attribute__((address_space(3))) v4i* lp =
        (__attribute__((address_space(3))) v4i*)(uintptr_t)(s + tid);
    __builtin_amdgcn_global_load_async_to_lds_b128(gp, lp, /*offset=*/0,
                                                   /*cpol=*/0);
  }
  __builtin_amdgcn_s_wait_asynccnt(0);
#else
  if (tid < TN) s[tid] = g[tid];
#endif
}

__device__ __forceinline__ float smooth_l1(float d) {
  float a = fabsf(d);
  return (a < 1.0f) ? (0.5f * a * a) : (a - 0.5f);
}

// ---------------------------------------------------------------------------
// Pass 0: zero the per-truth packed argmax array (ws is NOT re-zeroed by the
// harness between calls).
// ---------------------------------------------------------------------------
__global__ void k_init(unsigned long long* __restrict__ packed) {
  int i = blockIdx.x * blockDim.x + threadIdx.x;
  if (i < BN * TN) packed[i] = 0ull;
}

// ---------------------------------------------------------------------------
// Pass 1: IoU + per-anchor best (max/argmax over truths) + per-truth global
// best anchor via packed u64 atomicMax.  Grid (PN/BLK, BN), block BLK.
// ---------------------------------------------------------------------------
__global__ void k_iou(const float4* __restrict__ anchors,
                      const float4* __restrict__ targets,
                      float* __restrict__ bestOv,
                      int*   __restrict__ bestIdx,
                      unsigned long long* __restrict__ packed) {
  __shared__ float4 sT[TN];
  __shared__ float  sAT[TN];
  __shared__ float  sIou[32 * BLK];   // half the truths at a time (32 KB)

  const int tid = threadIdx.x;
  const int b   = blockIdx.y;
  const int p   = blockIdx.x * BLK + tid;

  stage_truths(targets + (size_t)b * TN, sT, tid);
  __syncthreads();
  if (tid < TN) {
    float4 t = sT[tid];
    sAT[tid] = (t.z - t.x) * (t.w - t.y);
  }

  const float4 A     = anchors[(size_t)b * PN + p];
  const float  areaA = (A.z - A.x) * (A.w - A.y);

  float best = -1.0f;
  int   bi   = 0;

  const int wave = tid >> 5, lane = tid & 31;

  for (int h = 0; h < 2; ++h) {
    __syncthreads();   // sT/sAT ready (h==0); previous half's reads done (h==1)
    #pragma unroll 8
    for (int tt = 0; tt < 32; ++tt) {
      const int    t  = h * 32 + tt;
      const float4 Tt = sT[t];
      float lx = fmaxf(Tt.x, A.x), ly = fmaxf(Tt.y, A.y);
      float rx = fminf(Tt.z, A.z), ry = fminf(Tt.w, A.w);
      float w  = fmaxf(rx - lx, 0.0f), hh = fmaxf(ry - ly, 0.0f);
      float inter = w * hh;
      float iou   = inter / (sAT[t] + areaA - inter);
      sIou[tt * BLK + tid] = iou;
      if (iou > best) { best = iou; bi = t; }   // strict > : first index wins
    }
    __syncthreads();
    // per-truth max over this block's 256 anchors: 8 waves x 4 truths each
    for (int tt = wave; tt < 32; tt += (BLK / 32)) {
      float m  = -1.0f;
      int   mk = 0;
      #pragma unroll
      for (int k = 0; k < BLK / 32; ++k) {
        int   idx = k * 32 + lane;
        float v   = sIou[tt * BLK + idx];
        if (v > m) { m = v; mk = idx; }
      }
      unsigned gp = (unsigned)(blockIdx.x * BLK + mk);
      // iou >= 0 so float bits are order-preserving; low word ~p -> ties pick
      // the smallest anchor index (matches jnp.argmax first-occurrence).
      unsigned long long key =
          ((unsigned long long)__float_as_uint(m) << 32) |
          (unsigned long long)(0xFFFFFFFFu - gp);
      #pragma unroll
      for (int off = 16; off > 0; off >>= 1) {
        unsigned long long o = __shfl_xor(key, off, 32);
        if (o > key) key = o;
      }
      if (lane == 0) atomicMax(&packed[b * TN + h * 32 + tt], key);
    }
  }

  bestOv[(size_t)b * PN + p]  = best;
  bestIdx[(size_t)b * PN + p] = bi;
}

// ---------------------------------------------------------------------------
// Pass 2: force-match each truth's best anchor (SSD match()).
// ---------------------------------------------------------------------------
__global__ void k_force(const unsigned long long* __restrict__ packed,
                        float* __restrict__ bestOv,
                        int*   __restrict__ bestIdx) {
  int i = blockIdx.x * blockDim.x + threadIdx.x;
  if (i >= BN * TN) return;
  int b = i / TN, t = i % TN;
  unsigned long long k = packed[i];
  unsigned p = 0xFFFFFFFFu - (unsigned)(k & 0xFFFFFFFFull);
  bestOv[(size_t)b * PN + p]  = 2.0f;
  bestIdx[(size_t)b * PN + p] = t;
}

// ---------------------------------------------------------------------------
// Pass 3: encode + smooth-L1 + cross-entropy; WMMA wave reduction to fixed
// per-block partials (deterministic, no float atomics).
// ---------------------------------------------------------------------------
__global__ void k_loss(const float4* __restrict__ anchors,
                       const float4* __restrict__ targets,
                       const float4* __restrict__ loc,
                       const float2* __restrict__ conf,
                       const float*  __restrict__ bestOv,
                       const int*    __restrict__ bestIdx,
                       float* __restrict__ partials) {
  __shared__ float4 sT[TN];
  __shared__ float  sW[(BLK / 32) * 3];

  const int tid = threadIdx.x;
  const int b   = blockIdx.y;
  const int p   = blockIdx.x * BLK + tid;

  stage_truths(targets + (size_t)b * TN, sT, tid);
  __syncthreads();

  const float ov  = bestOv[(size_t)b * PN + p];
  const int   ti  = bestIdx[(size_t)b * PN + p];
  const bool  pos = (ov >= 0.5f);

  const float4 A = anchors[(size_t)b * PN + p];
  const float4 M = sT[ti];

  float aw  = A.z - A.x,          ah  = A.w - A.y;
  float acx = (A.x + A.z) * 0.5f, acy = (A.y + A.w) * 0.5f;
  float mw  = M.z - M.x,          mh  = M.w - M.y;
  float mcx = (M.x + M.z) * 0.5f, mcy = (M.y + M.w) * 0.5f;

  float g0 = (mcx - acx) / (0.1f * aw);
  float g1 = (mcy - acy) / (0.1f * ah);
  float g2 = __logf(mw / aw) * 5.0f;   // / VAR_WH(=0.2)
  float g3 = __logf(mh / ah) * 5.0f;

  const float4 L = loc[(size_t)b * PN + p];
  float sl = smooth_l1(L.x - g0) + smooth_l1(L.y - g1) +
             smooth_l1(L.z - g2) + smooth_l1(L.w - g3);

  float slp = pos ? sl : 0.0f;
  float cnt = pos ? 1.0f : 0.0f;

  const float2 C  = conf[(size_t)b * PN + p];
  float mx  = fmaxf(C.x, C.y);
  float lse = mx + __logf(__expf(C.x - mx) + __expf(C.y - mx));
  float ce  = lse - (pos ? C.y : C.x);

  // Wave-level sums via WMMA (uniform control flow, EXEC all ones).
  float w0 = wave_sum32(slp);
  float w1 = wave_sum32(cnt);
  float w2 = wave_sum32(ce);

  const int wave = tid >> 5, lane = tid & 31;
  if (lane == 0) {
    sW[wave * 3 + 0] = w0;
    sW[wave * 3 + 1] = w1;
    sW[wave * 3 + 2] = w2;
  }
  __syncthreads();
  if (tid == 0) {
    float a0 = 0.f, a1 = 0.f, a2 = 0.f;
    #pragma unroll
    for (int w = 0; w < BLK / 32; ++w) {
      a0 += sW[w * 3 + 0];
      a1 += sW[w * 3 + 1];
      a2 += sW[w * 3 + 2];
    }
    int bid = blockIdx.y * gridDim.x + blockIdx.x;
    partials[bid * 3 + 0] = a0;
    partials[bid * 3 + 1] = a1;
    partials[bid * 3 + 2] = a2;
  }
}

// ---------------------------------------------------------------------------
// Pass 4: single-wave fixed-order reduction of block partials + final scalar.
// loss = loss_l / cnt  +  sum(ce) / (P * B * P)   (replicates reference).
// ---------------------------------------------------------------------------
__global__ void k_final(const float* __restrict__ partials,
                        float* __restrict__ out) {
  const int lane = threadIdx.x;   // launched with exactly 32 threads
  float s0 = 0.f, s1 = 0.f, s2 = 0.f;
  for (int i = lane; i < NBLK; i += 32) {
    s0 += partials[i * 3 + 0];
    s1 += partials[i * 3 + 1];
    s2 += partials[i * 3 + 2];
  }
  s0 = wave_sum32(s0);
  s1 = wave_sum32(s1);
  s2 = wave_sum32(s2);
  if (lane == 0) {
    const float inv = 1.0f / ((float)PN * (float)BN * (float)PN);
    out[0] = s0 / s1 + s2 * inv;
  }
}

// ---------------------------------------------------------------------------
extern "C" void kernel_launch(void* const* d_in, const int* in_sizes, int n_in,
                              void* d_out, int out_size, void* d_ws, size_t ws_size,
                              hipStream_t stream) {
  (void)in_sizes; (void)n_in; (void)out_size; (void)ws_size;

  const float4* loc     = (const float4*)d_in[0];   // [B,P,4] f32
  const float2* conf    = (const float2*)d_in[1];   // [B,P,2] f32
  const float4* anchors = (const float4*)d_in[2];   // [B,P,4] f32
  const float4* targets = (const float4*)d_in[3];   // [B,T,4] f32

  char* ws = (char*)d_ws;
  float*              bestOv   = (float*)ws;                                 // B*P f32
  int*                bestIdx  = (int*)(ws + (size_t)BN * PN * 4);           // B*P i32
  unsigned long long* packed   = (unsigned long long*)(ws + 2ull * BN * PN * 4);
  float*              partials = (float*)(ws + 2ull * BN * PN * 4 + (size_t)BN * TN * 8);

  dim3 gridP(PN / BLK, BN);

  hipLaunchKernelGGL(k_init, dim3((BN * TN + 255) / 256), dim3(256), 0, stream,
                     packed);
  hipLaunchKernelGGL(k_iou, gridP, dim3(BLK), 0, stream,
                     anchors, targets, bestOv, bestIdx, packed);
  hipLaunchKernelGGL(k_force, dim3((BN * TN + 255) / 256), dim3(256), 0, stream,
                     packed, bestOv, bestIdx);
  hipLaunchKernelGGL(k_loss, gridP, dim3(BLK), 0, stream,
                     anchors, targets, loc, conf, bestOv, bestIdx, partials);
  hipLaunchKernelGGL(k_final, dim3(1), dim3(32), 0, stream,
                     partials, (float*)d_out);
}